// TemporalGraphConvolutionalLayers_34711925686418
// MI455X (gfx1250) — compile-verified
//
#include <hip/hip_runtime.h>

#define T_STEPS 8
#define NNODES  20000
#define NEDGES  320000
#define CIN     64
#define HDIM    128
#define LN_EPS  1e-5f

typedef float v2f __attribute__((ext_vector_type(2)));
typedef float v8f __attribute__((ext_vector_type(8)));
typedef unsigned int u32x4 __attribute__((ext_vector_type(4)));
typedef int i32x8 __attribute__((ext_vector_type(8)));
typedef int i32x4 __attribute__((ext_vector_type(4)));

// ---------------------------------------------------------------- utilities
__global__ void fill_kernel(float* __restrict__ p, float v, int n) {
    int i = blockIdx.x * blockDim.x + threadIdx.x;
    if (i < n) p[i] = v;
}

__global__ void deg_accum_kernel(const long long* __restrict__ dst,
                                 const float* __restrict__ ew,
                                 float* __restrict__ deg, int E) {
    int e = blockIdx.x * blockDim.x + threadIdx.x;
    if (e < E) atomicAdd(&deg[dst[e]], ew[e]);
}

__global__ void dinv_kernel(float* __restrict__ dinv,
                            float* __restrict__ normSelf, int n) {
    int i = blockIdx.x * blockDim.x + threadIdx.x;
    if (i < n) {
        float d = dinv[i];
        float r = (d > 0.0f) ? rsqrtf(fmaxf(d, 1e-12f)) : 0.0f;
        dinv[i] = r;
        normSelf[i] = r * r;
    }
}

__global__ void edge_norm_kernel(const long long* __restrict__ src,
                                 const long long* __restrict__ dst,
                                 const float* __restrict__ ew,
                                 const float* __restrict__ dinv,
                                 float* __restrict__ norm, int E) {
    int e = blockIdx.x * blockDim.x + threadIdx.x;
    if (e < E) norm[e] = dinv[src[e]] * ew[e] * dinv[dst[e]];
}

// -------------------------------------------------- TDM 2D tile -> LDS
// Builds a Tensor DMA Descriptor (D#) per CDNA5 ISA sec. 8 and issues
// TENSOR_LOAD_TO_LDS. data_size = 4 bytes. Tile: tile_d0 (contiguous) x
// tile_d1 rows with row stride `stride0` elements. Dense pack into LDS.
__device__ __forceinline__ void tdm_load_2d(unsigned long long gaddr,
                                            unsigned int lds_addr,
                                            unsigned int tensor_d0,
                                            unsigned int tensor_d1,
                                            unsigned int tile_d0,
                                            unsigned int tile_d1,
                                            unsigned long long stride0) {
    u32x4 g0;
    g0[0] = 1u;                                    // count=1, user descriptor
    g0[1] = lds_addr;                              // LDS byte address
    g0[2] = (unsigned int)gaddr;                   // global_addr[31:0]
    g0[3] = ((unsigned int)(gaddr >> 32) & 0x1FFFFFFu) | (2u << 30); // [56:32], type=2
    i32x8 g1;
    g1[0] = (int)(2u << 16);                       // data_size=2 (4 bytes)
    g1[1] = (int)((tensor_d0 & 0xFFFFu) << 16);    // tensor_dim0[15:0]
    g1[2] = (int)((tensor_d0 >> 16) | ((tensor_d1 & 0xFFFFu) << 16));
    g1[3] = (int)((tensor_d1 >> 16) | (tile_d0 << 16));
    g1[4] = (int)tile_d1;                          // tile_dim1 (tile_dim2 = 0)
    g1[5] = (int)(unsigned int)stride0;            // tensor_dim0_stride[31:0]
    g1[6] = (int)((unsigned int)(stride0 >> 32) & 0xFFFFu);
    g1[7] = 0;
    i32x4 z4 = {0, 0, 0, 0};
    i32x8 z8 = {0, 0, 0, 0, 0, 0, 0, 0};
    __builtin_amdgcn_tensor_load_to_lds(g0, g1, z4, z4, z8, 0);
}

// ------------------------------------------------------------- WMMA GEMM
// C[M x Nout] = A[M x K] @ B (+ bias), fp32 via V_WMMA_F32_16X16X4_F32.
// TRANSB=false: B stored (K x Nout) row-major.  TRANSB=true: B stored
// (Nout x K) row-major and we multiply by its transpose (GRU W.T).
// Block = 256 threads (8 waves) owning 32 C rows. Waves: rowTile = w>>2,
// colTile = w&3. Column groups of 64: one TDM DMA per group stages the
// weight tile in LDS; A tile stays LDS-resident across groups.
template <int K, bool TRANSB>
__global__ __launch_bounds__(256) void gemm_wmma_kernel(
        const float* __restrict__ A, const float* __restrict__ B,
        const float* __restrict__ bias, float* __restrict__ C, int Nout) {
    __shared__ float As[32][K + 1];
    __shared__ float Bs[64 * K];
    const int tid = threadIdx.x;
    const long long mBase = (long long)blockIdx.x * 32;

    for (int idx = tid; idx < 32 * K; idx += 256) {
        int r = idx / K;
        int c = idx - r * K;
        As[r][c] = A[(mBase + r) * K + c];
    }

    const int wave    = tid >> 5;
    const int lane    = tid & 31;
    const int half    = lane >> 4;  // 0: lanes 0-15, 1: lanes 16-31
    const int l       = lane & 15;
    const int rowTile = wave >> 2;  // 0..1
    const int colTile = wave & 3;   // 0..3
    const int groups  = Nout >> 6;  // Nout / 64

    const unsigned int bsLds = (unsigned int)(unsigned long long)(&Bs[0]);

    for (int g = 0; g < groups; ++g) {
        __syncthreads();  // As ready (g==0) / prior group's compute finished
        if (tid == 0) {
            if (TRANSB) {
                // 64 rows of W (Nout x K), each K contiguous -> Bs[64][K]
                tdm_load_2d((unsigned long long)(const void*)(B + (long long)g * 64 * K),
                            bsLds, (unsigned int)K, 64u, (unsigned int)K, 64u,
                            (unsigned long long)K);
            } else {
                // K rows x 64 cols of B (K x Nout) -> Bs[K][64]
                tdm_load_2d((unsigned long long)(const void*)(B + g * 64),
                            bsLds, (unsigned int)Nout, (unsigned int)K, 64u,
                            (unsigned int)K, (unsigned long long)Nout);
            }
            __builtin_amdgcn_s_wait_tensorcnt(0);
        }
        __syncthreads();  // Bs tile visible to all waves

        const int n0 = g * 64 + colTile * 16;
        v8f acc = {};
        for (int k = 0; k < K; k += 4) {
            v2f a, b;
            // A fragment (16x4 f32): lanes 0-15 -> K=k,k+1 ; 16-31 -> K=k+2,k+3
            a.x = As[rowTile * 16 + l][k + 2 * half];
            a.y = As[rowTile * 16 + l][k + 2 * half + 1];
            if (TRANSB) {
                b.x = Bs[(colTile * 16 + l) * K + (k + 2 * half)];
                b.y = Bs[(colTile * 16 + l) * K + (k + 2 * half) + 1];
            } else {
                b.x = Bs[(k + 2 * half) * 64 + colTile * 16 + l];
                b.y = Bs[(k + 2 * half + 1) * 64 + colTile * 16 + l];
            }
            acc = __builtin_amdgcn_wmma_f32_16x16x4_f32(
                false, a, false, b, (short)0, acc, false, false);
        }
        float bv = bias ? bias[n0 + l] : 0.0f;
#pragma unroll
        for (int i = 0; i < 8; ++i) {
            long long row = mBase + rowTile * 16 + i + 8 * half;
            C[row * Nout + (n0 + l)] = acc[i] + bv;
        }
    }
}

// ------------------------------------------------- GCN scatter (segment_sum)
__global__ void init_s_kernel(const float* __restrict__ xt,
                              const float* __restrict__ normSelf,
                              const float* __restrict__ bg,
                              float* __restrict__ s) {
    int i = blockIdx.x * blockDim.x + threadIdx.x;
    if (i < NNODES * HDIM) {
        int n = i >> 7;
        int h = i & (HDIM - 1);
        s[i] = bg[h] + xt[i] * normSelf[n];
    }
}

__global__ void scatter_kernel(const float* __restrict__ xt,
                               const long long* __restrict__ src,
                               const long long* __restrict__ dst,
                               const float* __restrict__ norm,
                               float* __restrict__ s) {
    int idx = blockIdx.x * blockDim.x + threadIdx.x;  // E * (H/4) threads
    int e  = idx >> 5;
    int c4 = (idx & 31) << 2;
    if (e >= NEDGES) return;
    float w = norm[e];
    const float4 v = *(const float4*)(xt + src[e] * HDIM + c4);
    float* o = s + dst[e] * HDIM + c4;
    atomicAdd(o + 0, v.x * w);
    atomicAdd(o + 1, v.y * w);
    atomicAdd(o + 2, v.z * w);
    atomicAdd(o + 3, v.w * w);
}

// ------------------------------------------------- GRU gates + LayerNorm
__global__ void gru_ln_kernel(const float* __restrict__ gi,
                              const float* __restrict__ gh,
                              float* __restrict__ h,
                              const float* __restrict__ gamma,
                              const float* __restrict__ beta) {
    const int n = blockIdx.x;
    const int j = threadIdx.x;  // 0..127
    const float* gin = gi + (long long)n * (3 * HDIM);
    const float* ghn = gh + (long long)n * (3 * HDIM);

    float ir = gin[j], iz = gin[HDIM + j], in_ = gin[2 * HDIM + j];
    float hr = ghn[j], hz = ghn[HDIM + j], hn  = ghn[2 * HDIM + j];
    float r  = 1.0f / (1.0f + __expf(-(ir + hr)));
    float z  = 1.0f / (1.0f + __expf(-(iz + hz)));
    float nn = tanhf(in_ + r * hn);
    float hp = h[(long long)n * HDIM + j];
    float hv = (1.0f - z) * nn + z * hp;

    __shared__ float sbuf[HDIM];
    sbuf[j] = hv;
    __syncthreads();
    for (int off = 64; off; off >>= 1) {
        if (j < off) sbuf[j] += sbuf[j + off];
        __syncthreads();
    }
    float mu = sbuf[0] * (1.0f / HDIM);
    __syncthreads();
    float d = hv - mu;
    sbuf[j] = d * d;
    __syncthreads();
    for (int off = 64; off; off >>= 1) {
        if (j < off) sbuf[j] += sbuf[j + off];
        __syncthreads();
    }
    float var = sbuf[0] * (1.0f / HDIM);
    h[(long long)n * HDIM + j] = d * rsqrtf(var + LN_EPS) * gamma[j] + beta[j];
}

// ---------------------------------------------------------------- driver
extern "C" void kernel_launch(void* const* d_in, const int* in_sizes, int n_in,
                              void* d_out, int out_size, void* d_ws, size_t ws_size,
                              hipStream_t stream) {
    const float*     x_seq = (const float*)d_in[0];
    const long long* ei    = (const long long*)d_in[1];
    const float*     ew    = (const float*)d_in[2];
    const float* gcn_W[2] = {(const float*)d_in[3],  (const float*)d_in[11]};
    const float* gcn_b[2] = {(const float*)d_in[4],  (const float*)d_in[12]};
    const float* W_ih[2]  = {(const float*)d_in[5],  (const float*)d_in[13]};
    const float* W_hh[2]  = {(const float*)d_in[6],  (const float*)d_in[14]};
    const float* b_ih[2]  = {(const float*)d_in[7],  (const float*)d_in[15]};
    const float* b_hh[2]  = {(const float*)d_in[8],  (const float*)d_in[16]};
    const float* gamma[2] = {(const float*)d_in[9],  (const float*)d_in[17]};
    const float* beta[2]  = {(const float*)d_in[10], (const float*)d_in[18]};
    const long long* src = ei;
    const long long* dst = ei + NEDGES;

    const long long NH = (long long)NNODES * HDIM;
    float* ws       = (float*)d_ws;
    float* dinv     = ws;                 // N (deg first, then deg^-1/2)
    float* normSelf = dinv + NNODES;      // N
    float* norm     = normSelf + NNODES;  // E
    float* h0       = norm + NEDGES;      // N*H
    float* h1       = h0 + NH;            // N*H
    float* xt       = h1 + NH;            // N*H
    float* s        = xt + NH;            // N*H
    float* gi       = s + NH;             // N*3H
    float* gh       = gi + 3 * NH;        // N*3H

    // --- normalization precompute ------------------------------------
    fill_kernel<<<(NNODES + 255) / 256, 256, 0, stream>>>(dinv, 1.0f, NNODES);
    deg_accum_kernel<<<(NEDGES + 255) / 256, 256, 0, stream>>>(dst, ew, dinv, NEDGES);
    dinv_kernel<<<(NNODES + 255) / 256, 256, 0, stream>>>(dinv, normSelf, NNODES);
    edge_norm_kernel<<<(NEDGES + 255) / 256, 256, 0, stream>>>(src, dst, ew, dinv, norm, NEDGES);
    (void)hipMemsetAsync(h0, 0, NH * sizeof(float), stream);
    (void)hipMemsetAsync(h1, 0, NH * sizeof(float), stream);

    const int MB = NNODES / 32;                 // 625 blocks of 32 rows
    const int nhBlocks = (NNODES * HDIM + 255) / 256;
    const int scBlocks = (NEDGES * (HDIM / 4) + 255) / 256;

    for (int t = 0; t < T_STEPS; ++t) {
        // ----- layer 0: x_seq[t] -> h0 -----
        gemm_wmma_kernel<CIN, false><<<MB, 256, 0, stream>>>(
            x_seq + (long long)t * NNODES * CIN, gcn_W[0], nullptr, xt, HDIM);
        init_s_kernel<<<nhBlocks, 256, 0, stream>>>(xt, normSelf, gcn_b[0], s);
        scatter_kernel<<<scBlocks, 256, 0, stream>>>(xt, src, dst, norm, s);
        gemm_wmma_kernel<HDIM, true><<<MB, 256, 0, stream>>>(s,  W_ih[0], b_ih[0], gi, 3 * HDIM);
        gemm_wmma_kernel<HDIM, true><<<MB, 256, 0, stream>>>(h0, W_hh[0], b_hh[0], gh, 3 * HDIM);
        gru_ln_kernel<<<NNODES, HDIM, 0, stream>>>(gi, gh, h0, gamma[0], beta[0]);

        // ----- layer 1: h0 (this step's layer-0 output) -> h1 -----
        gemm_wmma_kernel<HDIM, false><<<MB, 256, 0, stream>>>(h0, gcn_W[1], nullptr, xt, HDIM);
        init_s_kernel<<<nhBlocks, 256, 0, stream>>>(xt, normSelf, gcn_b[1], s);
        scatter_kernel<<<scBlocks, 256, 0, stream>>>(xt, src, dst, norm, s);
        gemm_wmma_kernel<HDIM, true><<<MB, 256, 0, stream>>>(s,  W_ih[1], b_ih[1], gi, 3 * HDIM);
        gemm_wmma_kernel<HDIM, true><<<MB, 256, 0, stream>>>(h1, W_hh[1], b_hh[1], gh, 3 * HDIM);
        gru_ln_kernel<<<NNODES, HDIM, 0, stream>>>(gi, gh, h1, gamma[1], beta[1]);
    }

    // outputs: [seq[-1] (== h1 final), h0 final, h1 final]
    float* out = (float*)d_out;
    (void)hipMemcpyAsync(out,          h1, NH * sizeof(float), hipMemcpyDeviceToDevice, stream);
    (void)hipMemcpyAsync(out + NH,     h0, NH * sizeof(float), hipMemcpyDeviceToDevice, stream);
    (void)hipMemcpyAsync(out + 2 * NH, h1, NH * sizeof(float), hipMemcpyDeviceToDevice, stream);
}